// GraphLearner_2250562863217
// MI455X (gfx1250) — compile-verified
//
#include <hip/hip_runtime.h>
#include <math.h>

typedef __attribute__((ext_vector_type(2))) float v2f;
typedef __attribute__((ext_vector_type(8))) float v8f;

#define C_ 256
#define D_ 512

// ---------------------------------------------------------------------------
// WMMA GEMM, C[M,N] = A[M,K] * B[N,K]^T (+ per-column bias), all f32 row-major.
// One wave per 16x16 tile; V_WMMA_F32_16X16X4_F32, K stepped by 4.
// A frag: lane(l): M = l&15, K = k0 + 2*(l>>4) + j  (contiguous -> b64 load)
// B frag (of B^T): N = l&15, K = k0 + 2*(l>>4) + j -> B[N][K] contiguous too.
// C/D: VGPR v -> M = v + 8*(l>>4), N = l&15.
// ---------------------------------------------------------------------------
__global__ __launch_bounds__(128) void gemm_nt_wmma(
    const float* __restrict__ A, const float* __restrict__ B,
    const float* __restrict__ colBias, float* __restrict__ C,
    int M, int N, int K)
{
    const int lane  = threadIdx.x & 31;
    const int wave  = threadIdx.x >> 5;
    const int tilesN = N >> 4;
    const int t  = blockIdx.x * 4 + wave;
    const int m0 = (t / tilesN) << 4;
    const int n0 = (t % tilesN) << 4;
    const int half = lane >> 4;
    const int lm   = lane & 15;

    const float* arow = A + (size_t)(m0 + lm) * K + 2 * half;
    const float* brow = B + (size_t)(n0 + lm) * K + 2 * half;

    v8f acc = {};
    for (int k = 0; k < K; k += 4) {
        v2f a = *(const v2f*)(arow + k);
        v2f b = *(const v2f*)(brow + k);
        acc = __builtin_amdgcn_wmma_f32_16x16x4_f32(
            false, a, false, b, (short)0, acc, false, false);
    }

    const float bias = colBias ? colBias[n0 + lm] : 0.0f;
#pragma unroll
    for (int v = 0; v < 8; ++v)
        C[(size_t)(m0 + v + 8 * half) * N + n0 + lm] = acc[v] + bias;
}

// ---------------------------------------------------------------------------
// WMMA GEMM, Out = [tanh]( A[M,K] * B[K,N] + Cin + biasMat ), f32 row-major.
// ---------------------------------------------------------------------------
__global__ __launch_bounds__(128) void gemm_nn_wmma(
    const float* __restrict__ A, const float* __restrict__ B,
    const float* __restrict__ Cin, const float* __restrict__ biasMat,
    float* __restrict__ Out, int M, int N, int K, int doTanh)
{
    const int lane  = threadIdx.x & 31;
    const int wave  = threadIdx.x >> 5;
    const int tilesN = N >> 4;
    const int t  = blockIdx.x * 4 + wave;
    const int m0 = (t / tilesN) << 4;
    const int n0 = (t % tilesN) << 4;
    const int half = lane >> 4;
    const int lm   = lane & 15;

    const float* arow = A + (size_t)(m0 + lm) * K + 2 * half;
    const float* bcol = B + (size_t)(2 * half) * N + (n0 + lm);

    v8f acc = {};
    for (int k = 0; k < K; k += 4) {
        v2f a = *(const v2f*)(arow + k);
        v2f b;
        b[0] = bcol[(size_t)k * N];
        b[1] = bcol[(size_t)(k + 1) * N];
        acc = __builtin_amdgcn_wmma_f32_16x16x4_f32(
            false, a, false, b, (short)0, acc, false, false);
    }

#pragma unroll
    for (int v = 0; v < 8; ++v) {
        size_t idx = (size_t)(m0 + v + 8 * half) * N + n0 + lm;
        float x = acc[v];
        if (Cin)     x += Cin[idx];
        if (biasMat) x += biasMat[idx];
        Out[idx] = doTanh ? tanhf(x) : x;
    }
}

// ---------------------------------------------------------------------------
// Row L2-normalize: Xn[r] = X[r] / max(||X[r]||, 1e-12)
// ---------------------------------------------------------------------------
__global__ __launch_bounds__(128) void rownorm_kernel(
    const float* __restrict__ X, float* __restrict__ Xn, int D)
{
    __shared__ float red[128];
    const int row = blockIdx.x, tid = threadIdx.x;
    const float* x = X + (size_t)row * D;
    float s = 0.0f;
    for (int i = tid; i < D; i += 128) { float v = x[i]; s += v * v; }
    red[tid] = s; __syncthreads();
    for (int off = 64; off > 0; off >>= 1) {
        if (tid < off) red[tid] += red[tid + off];
        __syncthreads();
    }
    const float inv = 1.0f / fmaxf(sqrtf(red[0]), 1e-12f);
    float* y = Xn + (size_t)row * D;
    for (int i = tid; i < D; i += 128) y[i] = x[i] * inv;
}

// R[row] = sum_k max(G[row][k], 0)
__global__ __launch_bounds__(128) void clip_rowsum_kernel(
    const float* __restrict__ G, float* __restrict__ R, int N)
{
    __shared__ float red[128];
    const int row = blockIdx.x, tid = threadIdx.x;
    const float* g = G + (size_t)row * N;
    float s = 0.0f;
    for (int i = tid; i < N; i += 128) s += fmaxf(g[i], 0.0f);
    red[tid] = s; __syncthreads();
    for (int off = 64; off > 0; off >>= 1) {
        if (tid < off) red[tid] += red[tid + off];
        __syncthreads();
    }
    if (tid == 0) R[row] = red[0];
}

// ---------------------------------------------------------------------------
// Column-wise softmax prep. For column p: m = max over rows of S (and S0 if
// present); S := exp(S-m) in place (S0 likewise); T[p] = sum of exp'd S rows.
// ---------------------------------------------------------------------------
__global__ __launch_bounds__(128) void softmax_prep_kernel(
    float* __restrict__ S, float* __restrict__ S0,
    float* __restrict__ T, int rows, int cols)
{
    __shared__ float red[128];
    const int p = blockIdx.x, tid = threadIdx.x;
    float m = -3.0e38f;
    for (int r = tid; r < rows; r += 128) m = fmaxf(m, S[(size_t)r * cols + p]);
    if (S0)
        for (int r = tid; r < rows; r += 128) m = fmaxf(m, S0[(size_t)r * cols + p]);
    red[tid] = m; __syncthreads();
    for (int off = 64; off > 0; off >>= 1) {
        if (tid < off) red[tid] = fmaxf(red[tid], red[tid + off]);
        __syncthreads();
    }
    m = red[0];
    __syncthreads();
    float s = 0.0f;
    for (int r = tid; r < rows; r += 128) {
        size_t i = (size_t)r * cols + p;
        float e = expf(S[i] - m);
        S[i] = e; s += e;
    }
    if (S0)
        for (int r = tid; r < rows; r += 128) {
            size_t i = (size_t)r * cols + p;
            S0[i] = expf(S0[i] - m);
        }
    red[tid] = s; __syncthreads();
    for (int off = 64; off > 0; off >>= 1) {
        if (tid < off) red[tid] += red[tid + off];
        __syncthreads();
    }
    if (tid == 0) T[p] = red[0];
}

// ---------------------------------------------------------------------------
// Normalized adjacency row 0:
//   d0 = 2 + R0[b];  e = clip(Erow[b][k]);  dk = e + Rnode[k] + 1
//   aK[b][k] = e / sqrt(d0*dk);  a0[b] = 2/d0
// ---------------------------------------------------------------------------
__global__ __launch_bounds__(256) void adjacency_kernel(
    const float* __restrict__ Erow, const float* __restrict__ Rnode,
    const float* __restrict__ R0, float* __restrict__ a0,
    float* __restrict__ aK, int N)
{
    const int b = blockIdx.x, k = threadIdx.x;
    const float d0 = 2.0f + R0[b];
    const float e  = fmaxf(Erow[(size_t)b * N + k], 0.0f);
    const float dk = e + Rnode[k] + 1.0f;
    aK[(size_t)b * N + k] = e / sqrtf(d0 * dk);
    if (k == 0) a0[b] = 2.0f / d0;
}

// g[b][p] = (a0[b]*E0[b][p] + gnum[b][p]) / (E0[b][p] + T[p])   (in place)
__global__ __launch_bounds__(256) void g_finalize_kernel(
    float* __restrict__ g, const float* __restrict__ E0,
    const float* __restrict__ T, const float* __restrict__ a0, int N)
{
    const int b = blockIdx.x, p = threadIdx.x;
    const size_t i = (size_t)b * N + p;
    const float e0 = E0[i];
    g[i] = (a0[b] * e0 + g[i]) / (e0 + T[p]);
}

// V[b][:] += a0[b] * X[b][:]
__global__ __launch_bounds__(256) void axpy_row_kernel(
    float* __restrict__ V, const float* __restrict__ X,
    const float* __restrict__ a0, int D)
{
    const int b = blockIdx.x;
    const float a = a0[b];
    for (int d = threadIdx.x; d < D; d += blockDim.x)
        V[(size_t)b * D + d] += a * X[(size_t)b * D + d];
}

// main = 0.5*base_text + 0.5*(0.7*o_tt + 0.3*o_it)
__global__ __launch_bounds__(256) void combine_kernel(
    const float* __restrict__ bt, const float* __restrict__ ott,
    const float* __restrict__ oit, float* __restrict__ mo, int n)
{
    const int i = blockIdx.x * 256 + threadIdx.x;
    if (i < n) mo[i] = 0.5f * bt[i] + 0.5f * (0.7f * ott[i] + 0.3f * oit[i]);
}

__global__ __launch_bounds__(256) void copy_kernel(
    const float* __restrict__ src, float* __restrict__ dst, int n)
{
    const int i = blockIdx.x * 256 + threadIdx.x;
    if (i < n) dst[i] = src[i];
}

// ---------------------------------------------------------------------------
extern "C" void kernel_launch(void* const* d_in, const int* in_sizes, int n_in,
                              void* d_out, int out_size, void* d_ws, size_t ws_size,
                              hipStream_t stream)
{
    (void)in_sizes; (void)n_in; (void)out_size; (void)ws_size;
    const float* img_feature = (const float*)d_in[0];
    const float* base_text   = (const float*)d_in[1];
    const float* base_img    = (const float*)d_in[2];
    const float* base_neg    = (const float*)d_in[3];
    const float* p_list_t    = (const float*)d_in[4];
    const float* a_w_t       = (const float*)d_in[5];
    const float* a_b_t       = (const float*)d_in[6];
    const float* p_list_i    = (const float*)d_in[7];
    const float* a_w_i       = (const float*)d_in[8];
    const float* a_b_i       = (const float*)d_in[9];
    const float* W_tt        = (const float*)d_in[10];
    const float* b_tt        = (const float*)d_in[11];
    const float* W_it        = (const float*)d_in[12];
    const float* b_it        = (const float*)d_in[13];
    const float* W_ntt       = (const float*)d_in[14];
    const float* b_ntt       = (const float*)d_in[15];

    const int C = C_, D = D_;
    const int CD = C * D, CC = C * C;

    float* ws  = (float*)d_ws;
    float* Tn  = ws;            float* In  = Tn  + CD;  float* Nn  = In  + CD;
    float* Gt  = Nn  + CD;      float* Gi  = Gt  + CC;  float* Gn  = Gi  + CC;
    float* Cti = Gn  + CC;      float* Ctn = Cti + CC;
    float* Et  = Ctn + CC;      // St -> exp'd in place
    float* E0i = Et  + CC;      // Sti -> exp'd in place
    float* Ei  = E0i + CC;      // Sii -> exp'd in place
    float* Tt  = Ei  + CC;      float* Ti  = Tt + C;
    float* Rt  = Ti  + C;       float* Ri  = Rt + C;    float* Rn  = Ri + C;
    float* RCti= Rn  + C;       float* RCtn= RCti + C;
    float* a0t = RCtn+ C;       float* a0i = a0t + C;   float* a0n = a0i + C;
    float* aKt = a0n + C;       float* aKi = aKt + CC;  float* aKn = aKi + CC;
    float* g_t = aKn + CC;      float* g_i = g_t + CC;  // gnum -> g in place
    float* v_t = g_i + CC;      float* v_i = v_t + CD;  float* v_n = v_i + CD;

    float* out    = (float*)d_out;
    float* o_main = out;
    float* o_tt   = out + (size_t)CD;
    float* o_it   = out + (size_t)2 * CD;
    float* o_ntt  = out + (size_t)3 * CD;
    float* o_img  = out + (size_t)4 * CD;

    // 1) row-normalize bases
    rownorm_kernel<<<C, 128, 0, stream>>>(base_text, Tn, D);
    rownorm_kernel<<<C, 128, 0, stream>>>(base_img,  In, D);
    rownorm_kernel<<<C, 128, 0, stream>>>(base_neg,  Nn, D);

    // 2) Gram / cross-Gram + attention scores (A * B^T via WMMA)
    const int gCC = (C / 16) * (C / 16) / 4;   // 64 blocks
    const int gCD = (C / 16) * (D / 16) / 4;   // 128 blocks
    gemm_nt_wmma<<<gCC, 128, 0, stream>>>(Tn, Tn, nullptr, Gt,  C, C, D);
    gemm_nt_wmma<<<gCC, 128, 0, stream>>>(In, In, nullptr, Gi,  C, C, D);
    gemm_nt_wmma<<<gCC, 128, 0, stream>>>(Nn, Nn, nullptr, Gn,  C, C, D);
    gemm_nt_wmma<<<gCC, 128, 0, stream>>>(Tn, In, nullptr, Cti, C, C, D);
    gemm_nt_wmma<<<gCC, 128, 0, stream>>>(Tn, Nn, nullptr, Ctn, C, C, D);
    gemm_nt_wmma<<<gCC, 128, 0, stream>>>(base_text, a_w_t, a_b_t, Et,  C, C, D);
    gemm_nt_wmma<<<gCC, 128, 0, stream>>>(base_text, a_w_i, a_b_i, E0i, C, C, D);
    gemm_nt_wmma<<<gCC, 128, 0, stream>>>(base_img,  a_w_i, a_b_i, Ei,  C, C, D);

    // 3) clipped row-sums (degrees)
    clip_rowsum_kernel<<<C, 128, 0, stream>>>(Gt,  Rt,   C);
    clip_rowsum_kernel<<<C, 128, 0, stream>>>(Gi,  Ri,   C);
    clip_rowsum_kernel<<<C, 128, 0, stream>>>(Gn,  Rn,   C);
    clip_rowsum_kernel<<<C, 128, 0, stream>>>(Cti, RCti, C);
    clip_rowsum_kernel<<<C, 128, 0, stream>>>(Ctn, RCtn, C);

    // 4) softmax numerators + column sums
    softmax_prep_kernel<<<C, 128, 0, stream>>>(Et, nullptr, Tt, C, C);
    softmax_prep_kernel<<<C, 128, 0, stream>>>(Ei, E0i,     Ti, C, C);

    // 5) normalized adjacency row 0 per branch
    adjacency_kernel<<<C, C, 0, stream>>>(Gt,  Rt, Rt,   a0t, aKt, C);
    adjacency_kernel<<<C, C, 0, stream>>>(Cti, Ri, RCti, a0i, aKi, C);
    adjacency_kernel<<<C, C, 0, stream>>>(Ctn, Rn, RCtn, a0n, aKn, C);

    // 6) g numerators (aK @ E) then finalize g
    gemm_nn_wmma<<<gCC, 128, 0, stream>>>(aKt, Et, nullptr, nullptr, g_t, C, C, C, 0);
    gemm_nn_wmma<<<gCC, 128, 0, stream>>>(aKi, Ei, nullptr, nullptr, g_i, C, C, C, 0);
    g_finalize_kernel<<<C, C, 0, stream>>>(g_t, Et,  Tt, a0t, C);
    g_finalize_kernel<<<C, C, 0, stream>>>(g_i, E0i, Ti, a0i, C);

    // 7) v = aK @ base (+ g @ p_list) (+ a0 * base_text)
    gemm_nn_wmma<<<gCD, 128, 0, stream>>>(aKt, base_text, nullptr, nullptr, v_t, C, D, C, 0);
    gemm_nn_wmma<<<gCD, 128, 0, stream>>>(aKi, base_img,  nullptr, nullptr, v_i, C, D, C, 0);
    gemm_nn_wmma<<<gCD, 128, 0, stream>>>(aKn, base_neg,  nullptr, nullptr, v_n, C, D, C, 0);
    gemm_nn_wmma<<<gCD, 128, 0, stream>>>(g_t, p_list_t, v_t, nullptr, v_t, C, D, C, 0);
    gemm_nn_wmma<<<gCD, 128, 0, stream>>>(g_i, p_list_i, v_i, nullptr, v_i, C, D, C, 0);
    axpy_row_kernel<<<C, 256, 0, stream>>>(v_t, base_text, a0t, D);
    axpy_row_kernel<<<C, 256, 0, stream>>>(v_i, base_text, a0i, D);
    axpy_row_kernel<<<C, 256, 0, stream>>>(v_n, base_text, a0n, D);

    // 8) GCN output: tanh(v @ W + bias)  (fused epilogue)
    const int gDD = (C / 16) * (D / 16) / 4;   // 128 blocks, K = 512
    gemm_nn_wmma<<<gDD, 128, 0, stream>>>(v_t, W_tt,  nullptr, b_tt,  o_tt,  C, D, D, 1);
    gemm_nn_wmma<<<gDD, 128, 0, stream>>>(v_i, W_it,  nullptr, b_it,  o_it,  C, D, D, 1);
    gemm_nn_wmma<<<gDD, 128, 0, stream>>>(v_n, W_ntt, nullptr, b_ntt, o_ntt, C, D, D, 1);

    // 9) main combine + img passthrough
    combine_kernel<<<(CD + 255) / 256, 256, 0, stream>>>(base_text, o_tt, o_it, o_main, CD);
    copy_kernel<<<(64 * D + 255) / 256, 256, 0, stream>>>(img_feature, o_img, 64 * D);
}